// rmsdLoss_66941360276212
// MI455X (gfx1250) — compile-verified
//
#include <hip/hip_runtime.h>
#include <math.h>

#define BROWS 2048
#define LROW  4087      // 4*1024 - 9
#define NSTEP 1021      // scan steps (K-3)
#define SPL   32        // steps per lane (32*32 = 1024 padded)
#define NPAD  1024
#define CEPS  1e-6f

typedef unsigned int v4u __attribute__((ext_vector_type(4)));
typedef int          v4i __attribute__((ext_vector_type(4)));
typedef int          v8i __attribute__((ext_vector_type(8)));

struct Xf { float r[9]; float p[3]; };   // affine: R (row-major 3x3), p

// C = A * B  (homogeneous product: R = Ra*Rb, p = Ra*pb + pa)
__device__ __forceinline__ Xf xmul(const Xf& A, const Xf& B) {
  Xf C;
#pragma unroll
  for (int i = 0; i < 3; ++i) {
    float a0 = A.r[i*3+0], a1 = A.r[i*3+1], a2 = A.r[i*3+2];
#pragma unroll
    for (int j = 0; j < 3; ++j)
      C.r[i*3+j] = a0*B.r[0*3+j] + a1*B.r[1*3+j] + a2*B.r[2*3+j];
    C.p[i] = a0*B.p[0] + a1*B.p[1] + a2*B.p[2] + A.p[i];
  }
  return C;
}

__device__ __forceinline__ Xf xshfl_up(const Xf& x, int d) {
  Xf y;
#pragma unroll
  for (int i = 0; i < 9; ++i) y.r[i] = __shfl_up(x.r[i], (unsigned)d, 32);
#pragma unroll
  for (int i = 0; i < 3; ++i) y.p[i] = __shfl_up(x.p[i], (unsigned)d, 32);
  return y;
}

// Local step transform. Columns: bc' = (-ct, st*cx, st*sx), m' = (-st,-ct*cx,-ct*sx),
// n' = (0,-sx,cx); translation = l * bc'.
__device__ __forceinline__ Xf makeT(float ct, float st, float cx, float sx, float l) {
  Xf T;
  T.r[0] = -ct;    T.r[1] = -st;     T.r[2] = 0.f;
  T.r[3] = st*cx;  T.r[4] = -ct*cx;  T.r[5] = -sx;
  T.r[6] = st*sx;  T.r[7] = -ct*sx;  T.r[8] = cx;
  T.p[0] = -l*ct;  T.p[1] = l*st*cx; T.p[2] = l*st*sx;
  return T;
}

// transposed param layout: reader (lane k, step k*32+j) is bank-conflict-free
__device__ __forceinline__ int pidx(int s) {
  return (((s & 31) << 5) | (s >> 5)) * 5;
}

#if defined(__HIP_DEVICE_COMPILE__) && defined(__gfx1250__)
// One TDM descriptor: contiguous 1-D row of LROW f32 -> LDS
__device__ __forceinline__ void tdm_load_row(const float* src, float* ldsDst) {
  unsigned long long ga = (unsigned long long)(uintptr_t)src;
  unsigned lds_off = (unsigned)(uintptr_t)ldsDst;          // LDS aperture: low 32b = offset
  v4u g0 = { 1u,                                           // count = 1 valid descriptor
             lds_off,                                      // lds_addr
             (unsigned)(ga & 0xFFFFFFFFull),               // global_addr[31:0]
             (unsigned)(((ga >> 32) & 0x01FFFFFFull) | 0x80000000u) }; // type=2
  v8i g1 = { (int)0x00020000,                              // data_size = 4B, no multicast
             (int)((LROW & 0xFFFF) << 16),                 // tensor_dim0[15:0]
             (int)(((LROW >> 16) & 0xFFFF) | (1 << 16)),   // td0[31:16] | tensor_dim1 = 1
             (int)((LROW & 0xFFFF) << 16),                 // td1[31:16]=0 | tile_dim0 = LROW
             0,                                            // tile_dim1 = 0 (1-D), tile_dim2 = 0
             (int)LROW,                                    // tensor_dim0_stride[31:0]
             0, 0 };
  v4i gz4 = { 0, 0, 0, 0 };
  v8i gz8 = { 0, 0, 0, 0, 0, 0, 0, 0 };
  __builtin_amdgcn_tensor_load_to_lds(g0, g1, gz4, gz4, gz8, 0);
}
#endif

__global__ void __launch_bounds__(32)
rmsd_chains(const float* __restrict__ pred, const float* __restrict__ targ,
            float* __restrict__ out) {
  __shared__ float rowBufP[4096];     // 16 KB  predictions row
  __shared__ float rowBufT[4096];     // 16 KB  targets row
  __shared__ float prm[NPAD * 5];     // 20 KB  (ct, st, cx, sx, l) per step
  __shared__ float pc[3 * 1024];      // 12 KB  prediction coordinates

  const int row  = blockIdx.x;
  const int lane = threadIdx.x;
  float acc = 0.f;

  // ---- issue BOTH row DMAs up-front; TDM ops from one wave complete in order,
  //      so the targets transfer overlaps the whole predictions pass ----
#if defined(__HIP_DEVICE_COMPILE__) && defined(__gfx1250__)
  tdm_load_row(pred + (size_t)row * LROW, rowBufP);
  tdm_load_row(targ + (size_t)row * LROW, rowBufT);
#else
  for (int i = lane; i < LROW; i += 32) {
    rowBufP[i] = pred[(size_t)row * LROW + i];
    rowBufT[i] = targ[(size_t)row * LROW + i];
  }
#endif

  for (int pass = 0; pass < 2; ++pass) {
#if defined(__HIP_DEVICE_COMPILE__) && defined(__gfx1250__)
    if (pass == 0) __builtin_amdgcn_s_wait_tensorcnt(1);   // first DMA done (in-order)
    else           __builtin_amdgcn_s_wait_tensorcnt(0);   // both DMAs done
#endif
    __syncthreads();

    const float* rowBuf = (pass == 0) ? rowBufP : rowBufT;
    const bool doClip = (pass == 0);   // predictions are clipped, targets are not
    auto fetch = [&](int i) -> float {
      float v = rowBuf[i];
      return doClip ? fminf(fmaxf(v, -1.f), 1.f) : v;
    };
    auto dn13 = [&](int j) { return (fetch(2042 + j) + 1.f) * 0.5f * 3.5f + 1.5f; };
    auto dnbl = [&](int j) { return (fetch(3064 + j) + 1.f) * 0.5f * 2.1f + 0.9f; };

    // ---- decode per-step params (no arccos/atan2: only sqrt/rsqrt needed) ----
    for (int i = lane; i < NPAD; i += 32) {
      float ct, st, cx, sx, l;
      if (i < NSTEP) {
        float l1 = dnbl(i + 1), l2 = dnbl(i + 2), d = dn13(i + 1);
        ct = (l1 * l1 + l2 * l2 - d * d) / (2.f * l1 * l2);
        ct = fminf(fmaxf(ct, -1.f + CEPS), 1.f - CEPS);
        st = sqrtf(fmaxf(1.f - ct * ct, 0.f));
        float s0 = fetch(2 * i), s1 = fetch(2 * i + 1);
        float r2 = s0 * s0 + s1 * s1;
        if (r2 > 1e-30f) { float ri = rsqrtf(r2); cx = s1 * ri; sx = s0 * ri; }
        else             { cx = 1.f; sx = 0.f; }
        l = l2;
      } else {  // identity padding
        ct = -1.f; st = 0.f; cx = 1.f; sx = 0.f; l = 0.f;
      }
      int b = pidx(i);
      prm[b+0] = ct; prm[b+1] = st; prm[b+2] = cx; prm[b+3] = sx; prm[b+4] = l;
    }
    __syncthreads();

    // ---- initial frame from (a0, a1, a2) ----
    float bl0 = dnbl(0), bl1 = dnbl(1), d0 = dn13(0);
    float ct0 = (bl0 * bl0 + bl1 * bl1 - d0 * d0) / (2.f * bl0 * bl1);
    ct0 = fminf(fmaxf(ct0, -1.f + CEPS), 1.f - CEPS);
    float st0 = sqrtf(fmaxf(1.f - ct0 * ct0, 0.f));
    Xf F0;                                   // columns: bc0, m0, n0 ; p = a2
    F0.r[0] = -ct0; F0.r[1] = -st0; F0.r[2] = 0.f;
    F0.r[3] =  st0; F0.r[4] = -ct0; F0.r[5] = 0.f;
    F0.r[6] =  0.f; F0.r[7] =  0.f; F0.r[8] = 1.f;
    F0.p[0] = bl0 - bl1 * ct0; F0.p[1] = bl1 * st0; F0.p[2] = 0.f;

    auto loadT = [&](int s) -> Xf {
      int b = pidx(s);
      return makeT(prm[b], prm[b+1], prm[b+2], prm[b+3], prm[b+4]);
    };

    // ---- per-lane serial composition of 32 local transforms ----
    const int base = lane * SPL;
    Xf L = loadT(base);
    for (int j = 1; j < SPL; ++j) L = xmul(L, loadT(base + j));

    // ---- wave32 inclusive scan (Hillis-Steele) over lane products ----
    Xf S = L;
#pragma unroll
    for (int d = 1; d < 32; d <<= 1) {
      Xf t = xshfl_up(S, d);
      if (lane >= d) S = xmul(t, S);
    }
    Xf E = xshfl_up(S, 1);                      // exclusive prefix
    Xf G = (lane == 0) ? F0 : xmul(F0, E);      // global starting frame per lane

    // ---- first three coordinates (a0, a1, a2) ----
    if (lane == 0) {
      if (pass == 0) {
        pc[0] = 0.f;     pc[1] = 0.f;     pc[2] = 0.f;
        pc[3] = bl0;     pc[4] = 0.f;     pc[5] = 0.f;
        pc[6] = F0.p[0]; pc[7] = F0.p[1]; pc[8] = F0.p[2];
      } else {
        float d3 = pc[3] - bl0;
        float d6 = pc[6] - F0.p[0], d7 = pc[7] - F0.p[1], d8 = pc[8] - F0.p[2];
        acc += pc[0]*pc[0] + pc[1]*pc[1] + pc[2]*pc[2];
        acc += d3*d3 + pc[4]*pc[4] + pc[5]*pc[5];
        acc += d6*d6 + d7*d7 + d8*d8;
      }
    }

    // ---- replay: emit/compare coordinates ----
    Xf F = G;
    for (int j = 0; j < SPL; ++j) {
      int s = base + j;
      F = xmul(F, loadT(s));
      if (s < NSTEP) {
        int ci = (3 + s) * 3;
        if (pass == 0) {
          pc[ci] = F.p[0]; pc[ci+1] = F.p[1]; pc[ci+2] = F.p[2];
        } else {
          float dx = pc[ci] - F.p[0], dy = pc[ci+1] - F.p[1], dz = pc[ci+2] - F.p[2];
          acc += dx*dx + dy*dy + dz*dz;
        }
      }
    }
    __syncthreads();
  }

  // ---- wave reduction + scaled global accumulation (mean over B*K*3) ----
#pragma unroll
  for (int off = 16; off > 0; off >>= 1) acc += __shfl_down(acc, (unsigned)off, 32);
  if (lane == 0) atomicAdd(out, acc * (1.0f / 6291456.0f)); // 2048*1024*3
}

__global__ void zero_kernel(float* out) { out[0] = 0.f; }

extern "C" void kernel_launch(void* const* d_in, const int* in_sizes, int n_in,
                              void* d_out, int out_size, void* d_ws, size_t ws_size,
                              hipStream_t stream) {
  (void)in_sizes; (void)n_in; (void)out_size; (void)d_ws; (void)ws_size;
  const float* pred = (const float*)d_in[0];
  const float* targ = (const float*)d_in[1];
  float* out = (float*)d_out;
  zero_kernel<<<1, 1, 0, stream>>>(out);
  rmsd_chains<<<BROWS, 32, 0, stream>>>(pred, targ, out);
}